// PositiveMultiheadAttention_85779086835795
// MI455X (gfx1250) — compile-verified
//
#include <hip/hip_runtime.h>

// Problem constants (from reference)
#define DM   512   // d_model
#define NH   8     // heads
#define HD   64    // head dim
#define BH   16    // BATCH * NHEAD
#define SEQ  1024
#define NTOK 2048  // BATCH * SEQ
#define SPLIT 4    // key-dimension split for k_kv occupancy

// Blocked GEMM geometry
#define BM 128
#define BN 128
#define BK 32
#define LDST 36            // LDS row stride (floats): conflict-free & 16B-aligned
#define NSTAGE (DM / BK)   // 16

typedef float v2f __attribute__((ext_vector_type(2)));
typedef float v4f __attribute__((ext_vector_type(4)));
typedef float v8f __attribute__((ext_vector_type(8)));
typedef int   v4i __attribute__((ext_vector_type(4)));

#if __has_builtin(__builtin_amdgcn_global_load_async_to_lds_b128)
#define HAVE_ASYNC 1
#else
#define HAVE_ASYNC 0
#endif

// D = A(16x4 f32) * B(4x16 f32) + C, full f32 precision on the matrix pipe.
static __device__ __forceinline__ v8f wmma4(v2f a, v2f b, v8f c) {
    return __builtin_amdgcn_wmma_f32_16x16x4_f32(
        false, a, false, b, (short)0, c, false, false);
}

// 16-byte global -> LDS copy; async (ASYNCcnt) when the toolchain exposes it.
static __device__ __forceinline__ void cp16(const float* __restrict__ g, float* l) {
#if HAVE_ASYNC
    __builtin_amdgcn_global_load_async_to_lds_b128((v4i*)g, (v4i*)l, 0, 0);
#else
    *(v4f*)l = *(const v4f*)g;
#endif
}

static __device__ __forceinline__ void wait_async(bool more_inflight) {
#if HAVE_ASYNC
    if (more_inflight) asm volatile("s_wait_asynccnt 8" ::: "memory");
    else               asm volatile("s_wait_asynccnt 0" ::: "memory");
#endif
}

// Copy one BK-wide slice of both 128-row operands into LDS [128][LDST].
// 256 threads, 4 x b128 per operand per thread (8 async ops / wave / stage).
static __device__ __forceinline__ void stage_copy(const float* __restrict__ gA,
                                                  const float* __restrict__ gW,
                                                  float* As, float* Bs,
                                                  int k0, int tid) {
    const int r = tid >> 3;           // 0..31
    const int c = (tid & 7) * 4;      // 0..28
#pragma unroll
    for (int i = 0; i < 4; ++i) {
        const int row = r + 32 * i;
        cp16(gA + (size_t)row * DM + k0 + c, As + row * LDST + c);
        cp16(gW + (size_t)row * DM + k0 + c, Bs + row * LDST + c);
    }
}

// 128x128 output block of A[128 x 512] @ W^T, double-buffered through LDS.
// 8 waves in a 2x4 grid; each wave owns 4x2 16x16 WMMA tiles (acc[mt*2+nt]).
static __device__ __forceinline__ void gemm_block(
    const float* __restrict__ gA, const float* __restrict__ gW,
    float* As0, float* Bs0, float* As1, float* Bs1,
    int tid, v8f acc[8]) {
    const int lane = tid & 31, lid = lane & 15, kk = (lane >> 4) * 2;
    const int wid = tid >> 5, wm = wid >> 2, wn = wid & 3;
    stage_copy(gA, gW, As0, Bs0, 0, tid);
    for (int s = 0; s < NSTAGE; ++s) {
        const bool more = (s + 1 < NSTAGE);
        if (more)
            stage_copy(gA, gW, ((s + 1) & 1) ? As1 : As0,
                       ((s + 1) & 1) ? Bs1 : Bs0, (s + 1) * BK, tid);
        wait_async(more);      // previous stage landed in LDS
        __syncthreads();
        const float* As = (s & 1) ? As1 : As0;
        const float* Bs = (s & 1) ? Bs1 : Bs0;
#pragma unroll
        for (int ks = 0; ks < BK / 4; ++ks) {
            v2f af[4], bf[2];
#pragma unroll
            for (int mt = 0; mt < 4; ++mt)
                af[mt] = *(const v2f*)(As + (wm * 64 + mt * 16 + lid) * LDST + ks * 4 + kk);
#pragma unroll
            for (int nt = 0; nt < 2; ++nt)
                bf[nt] = *(const v2f*)(Bs + (wn * 32 + nt * 16 + lid) * LDST + ks * 4 + kk);
#pragma unroll
            for (int mt = 0; mt < 4; ++mt)
#pragma unroll
                for (int nt = 0; nt < 2; ++nt)
                    acc[mt * 2 + nt] = wmma4(af[mt], bf[nt], acc[mt * 2 + nt]);
        }
        __syncthreads();
    }
}

// ---------------- Kernel 1: fused Q/K/V projections -------------------------
// qkv_ws layout: [mat(3)][b*NH+h][l][d]  (head-major per matrix)
__global__ __launch_bounds__(256) void k_proj_qkv(
    const float* __restrict__ x,
    const float* __restrict__ WQ, const float* __restrict__ bQ,
    const float* __restrict__ WK, const float* __restrict__ bK,
    const float* __restrict__ WV, const float* __restrict__ bV,
    float* __restrict__ qkv_ws) {
    __shared__ __align__(16) float As0[BM * LDST], Bs0[BN * LDST];
    __shared__ __align__(16) float As1[BM * LDST], Bs1[BN * LDST];
    const int bx  = blockIdx.x;
    const int mat = bx / 64;            // (2048/128)*(512/128) = 64 blocks/matrix
    const int rem = bx % 64;
    const int m0  = (rem >> 2) * BM;
    const int n0  = (rem & 3) * BN;
    const float* W    = (mat == 0) ? WQ : (mat == 1) ? WK : WV;
    const float* bias = (mat == 0) ? bQ : (mat == 1) ? bK : bV;
    const int tid = threadIdx.x;

    v8f acc[8] = {};
    gemm_block(x + (size_t)m0 * DM, W + (size_t)n0 * DM,
               As0, Bs0, As1, Bs1, tid, acc);

    const int lane = tid & 31, half = lane >> 4, lid = lane & 15;
    const int wid = tid >> 5, wm = wid >> 2, wn = wid & 3;
    float* out = qkv_ws + (size_t)mat * NTOK * DM;
#pragma unroll
    for (int mt = 0; mt < 4; ++mt) {
#pragma unroll
        for (int nt = 0; nt < 2; ++nt) {
            const int n = n0 + wn * 32 + nt * 16 + lid;
            const int h = n >> 6, d = n & 63;
            const float bv = bias[n];
            const v8f a = acc[mt * 2 + nt];
#pragma unroll
            for (int r = 0; r < 8; ++r) {
                const int m = m0 + wm * 64 + mt * 16 + r + 8 * half;
                const int b = m >> 10, l = m & 1023;
                out[(((size_t)(b * NH + h)) * SEQ + l) * HD + d] = a[r] + bv;
            }
        }
    }
}

// ---------------- Kernel 2: per-head key reduction (split 4-way) ------------
// Each block: full-column mK (identical across splits), partial s[d] and
// partial KV+/- = exp(K-mK)^T @ relu(+/-V) over its 256-key slice.
__global__ __launch_bounds__(512) void k_kv(
    const float* __restrict__ qkv_ws,
    float* __restrict__ kvp, float* __restrict__ kvm,
    float* __restrict__ shat, float* __restrict__ mkg) {
    const int bh  = blockIdx.x >> 2;
    const int sp  = blockIdx.x & 3;
    const int j0s = sp * (SEQ / SPLIT);
    const int j1s = j0s + (SEQ / SPLIT);
    const float* K = qkv_ws + (size_t)NTOK * DM + (size_t)bh * SEQ * HD;
    const float* V = qkv_ws + 2 * (size_t)NTOK * DM + (size_t)bh * SEQ * HD;

    __shared__ float red[8 * 64];
    __shared__ float mk[64];

    const int tid = threadIdx.x;
    const int d = tid & 63, g = tid >> 6;

    // full-column max of K (redundant per split, bit-identical)
    float m = -3.4e38f;
    for (int j = g; j < SEQ; j += 8) m = fmaxf(m, K[j * HD + d]);
    red[g * 64 + d] = m;
    __syncthreads();
    if (tid < 64) {
        float mm = red[tid];
        for (int gg = 1; gg < 8; ++gg) mm = fmaxf(mm, red[gg * 64 + tid]);
        mk[tid] = mm;
    }
    __syncthreads();

    // partial s[d] over this split's keys
    const float mkd = mk[d];
    float s = 0.f;
    for (int j = j0s + g; j < j1s; j += 8) s += __expf(K[j * HD + d] - mkd);
    red[g * 64 + d] = s;
    __syncthreads();
    if (tid < 64) {
        float ss = 0.f;
        for (int gg = 0; gg < 8; ++gg) ss += red[gg * 64 + tid];
        shat[((size_t)sp * BH + bh) * 64 + tid] = ss;
        if (sp == 0) mkg[bh * 64 + tid] = mk[tid];
    }
    __syncthreads();

    // partial KV+/-: 16 waves own the 4x4 grid of 16x16 tiles
    const int wave = tid >> 5, lane = tid & 31, half = lane >> 4, lid = lane & 15;
    const int m0 = (wave >> 2) * 16;   // d tile
    const int n0 = (wave & 3) * 16;    // dv tile
    const int kko = half * 2;
    const float mka = mk[m0 + lid];
    const float* Kc = K + (m0 + lid);  // column view, stride HD over j
    const float* Vc = V + (n0 + lid);
    v8f accP = {}, accM = {};
    for (int j0 = j0s; j0 < j1s; j0 += 4) {
        if ((j0 & 31) == 0 && j0 + 32 < j1s) {   // uniform; global_prefetch_b8
            __builtin_prefetch(Kc + (size_t)(j0 + 32) * HD, 0, 1);
            __builtin_prefetch(Vc + (size_t)(j0 + 32) * HD, 0, 1);
        }
        const int j = j0 + kko;
        v2f a;  a.x = __expf(Kc[(size_t)j * HD] - mka);
                a.y = __expf(Kc[(size_t)(j + 1) * HD] - mka);
        const float v0 = Vc[(size_t)j * HD], v1 = Vc[(size_t)(j + 1) * HD];
        v2f bp; bp.x = fmaxf(v0, 0.f);  bp.y = fmaxf(v1, 0.f);
        v2f bm; bm.x = fmaxf(-v0, 0.f); bm.y = fmaxf(-v1, 0.f);
        accP = wmma4(a, bp, accP);   // shared A fragment feeds both
        accM = wmma4(a, bm, accM);
    }
    const size_t base = ((size_t)sp * BH + bh) * 4096;
#pragma unroll
    for (int r = 0; r < 8; ++r) {
        const int dd = m0 + r + 8 * half, dv = n0 + lid;
        kvp[base + dd * 64 + dv] = accP[r];
        kvm[base + dd * 64 + dv] = accM[r];
    }
}

// ---------------- Kernel 3: attention core (per 64 query rows) --------------
__global__ __launch_bounds__(512) void k_attn(
    const float* __restrict__ qkv_ws,
    const float* __restrict__ kvp, const float* __restrict__ kvm,
    const float* __restrict__ shat, const float* __restrict__ mkg,
    float* __restrict__ joined) {
    const int bh = blockIdx.x >> 4;
    const int rb = blockIdx.x & 15;
    const int b = bh >> 3, h = bh & 7;
    const int l0 = rb * 64;
    const float* Q = qkv_ws + (size_t)bh * SEQ * HD;

    __shared__ float eL[64 * 64];
    __shared__ float kvpL[64 * 64];
    __shared__ float kvmL[64 * 64];
    __shared__ float den[64];
    __shared__ float red[64 * 8];
    __shared__ float shl[64];
    __shared__ float mkl[64];

    const int tid = threadIdx.x;
    // stage KV while summing the 4 split-partials
    for (int i = tid; i < 4096; i += 512) {
        float p = 0.f, q = 0.f;
#pragma unroll
        for (int sp = 0; sp < SPLIT; ++sp) {
            p += kvp[((size_t)sp * BH + bh) * 4096 + i];
            q += kvm[((size_t)sp * BH + bh) * 4096 + i];
        }
        kvpL[i] = p; kvmL[i] = q;
    }
    if (tid < 64) {
        float s = 0.f;
#pragma unroll
        for (int sp = 0; sp < SPLIT; ++sp) s += shat[((size_t)sp * BH + bh) * 64 + tid];
        shl[tid] = s;
        mkl[tid] = mkg[bh * 64 + tid];
    }
    __syncthreads();

    // stabilized e = exp(Q + mK - rowmax); den = sum_d e * s[d]
    const int row = tid >> 3, part = tid & 7;    // 8 threads per row
    const float* qrow = Q + (size_t)(l0 + row) * HD + part * 8;
    float qv[8];
    float m = -3.4e38f;
#pragma unroll
    for (int i = 0; i < 8; ++i) {
        qv[i] = qrow[i] + mkl[part * 8 + i];
        m = fmaxf(m, qv[i]);
    }
    red[row * 8 + part] = m;
    __syncthreads();
    float rm = red[row * 8];
#pragma unroll
    for (int i = 1; i < 8; ++i) rm = fmaxf(rm, red[row * 8 + i]);
    __syncthreads();    // red about to be reused
    float ds = 0.f;
#pragma unroll
    for (int i = 0; i < 8; ++i) {
        const float e = __expf(qv[i] - rm);
        eL[row * 64 + part * 8 + i] = e;
        ds += e * shl[part * 8 + i];
    }
    red[row * 8 + part] = ds;
    __syncthreads();
    if (part == 0) {
        float dd = 0.f;
#pragma unroll
        for (int i = 0; i < 8; ++i) dd += red[row * 8 + i];
        den[row] = dd;
    }
    __syncthreads();

    // A± = e @ KV±  (16 waves, 4x4 tiles of 16x16, K = 64)
    const int wave = tid >> 5, lane = tid & 31, half = lane >> 4, lid = lane & 15;
    const int m0 = (wave >> 2) * 16;   // local row tile
    const int n0 = (wave & 3) * 16;    // dv tile
    const int kko = half * 2;
    const float* eRow = eL + (m0 + lid) * 64 + kko;
    v8f accP = {}, accM = {};
#pragma unroll 4
    for (int d0 = 0; d0 < HD; d0 += 4) {
        v2f a;  a.x = eRow[d0];                          a.y = eRow[d0 + 1];
        v2f bp; bp.x = kvpL[(d0 + kko) * 64 + n0 + lid]; bp.y = kvpL[(d0 + kko + 1) * 64 + n0 + lid];
        v2f bm; bm.x = kvmL[(d0 + kko) * 64 + n0 + lid]; bm.y = kvmL[(d0 + kko + 1) * 64 + n0 + lid];
        accP = wmma4(a, bp, accP);
        accM = wmma4(a, bm, accM);
    }
#pragma unroll
    for (int r = 0; r < 8; ++r) {
        const int rr = m0 + r + 8 * half;   // local row 0..63
        const int dv = n0 + lid;
        const float val = (accP[r] - accM[r]) / den[rr];
        joined[((size_t)(b * SEQ + l0 + rr)) * DM + h * HD + dv] = val;
    }
}

// ---------------- Kernel 4: output projection -------------------------------
__global__ __launch_bounds__(256) void k_proj_out(
    const float* __restrict__ joined,
    const float* __restrict__ WO, const float* __restrict__ bO,
    float* __restrict__ out) {
    __shared__ __align__(16) float As0[BM * LDST], Bs0[BN * LDST];
    __shared__ __align__(16) float As1[BM * LDST], Bs1[BN * LDST];
    const int bx = blockIdx.x;           // 64 blocks
    const int m0 = (bx >> 2) * BM;
    const int n0 = (bx & 3) * BN;
    const int tid = threadIdx.x;

    v8f acc[8] = {};
    gemm_block(joined + (size_t)m0 * DM, WO + (size_t)n0 * DM,
               As0, Bs0, As1, Bs1, tid, acc);

    const int lane = tid & 31, half = lane >> 4, lid = lane & 15;
    const int wid = tid >> 5, wm = wid >> 2, wn = wid & 3;
#pragma unroll
    for (int mt = 0; mt < 4; ++mt) {
#pragma unroll
        for (int nt = 0; nt < 2; ++nt) {
            const int n = n0 + wn * 32 + nt * 16 + lid;
            const float bv = bO[n];
            const v8f a = acc[mt * 2 + nt];
#pragma unroll
            for (int r = 0; r < 8; ++r) {
                const int m = m0 + wm * 64 + mt * 16 + r + 8 * half;
                out[(size_t)m * DM + n] = a[r] + bv;
            }
        }
    }
}

extern "C" void kernel_launch(void* const* d_in, const int* in_sizes, int n_in,
                              void* d_out, int out_size, void* d_ws, size_t ws_size,
                              hipStream_t stream) {
    const float* x  = (const float*)d_in[0];
    const float* WQ = (const float*)d_in[1];
    const float* bQ = (const float*)d_in[2];
    const float* WK = (const float*)d_in[3];
    const float* bK = (const float*)d_in[4];
    const float* WV = (const float*)d_in[5];
    const float* bV = (const float*)d_in[6];
    const float* WO = (const float*)d_in[7];
    const float* bO = (const float*)d_in[8];

    float* ws     = (float*)d_ws;
    float* qkv    = ws;                                  // 3 * 2048*512
    float* kvp    = ws + (size_t)3 * NTOK * DM;          // SPLIT * 16 * 4096
    float* kvm    = kvp + (size_t)SPLIT * BH * 4096;
    float* shat   = kvm + (size_t)SPLIT * BH * 4096;     // SPLIT * 16 * 64
    float* mkg    = shat + (size_t)SPLIT * BH * 64;      // 16 * 64
    float* joined = mkg + BH * 64;                       // 2048*512

    k_proj_qkv<<<192, 256, 0, stream>>>(x, WQ, bQ, WK, bK, WV, bV, qkv);
    k_kv      <<<BH * SPLIT, 512, 0, stream>>>(qkv, kvp, kvm, shat, mkg);
    k_attn    <<<BH * 16, 512, 0, stream>>>(qkv, kvp, kvm, shat, mkg, joined);
    k_proj_out<<<64, 256, 0, stream>>>(joined, WO, bO, (float*)d_out);
}